// ProductKeyMemory_42606075576724
// MI455X (gfx1250) — compile-verified
//
#include <hip/hip_runtime.h>

typedef __attribute__((ext_vector_type(2))) float v2f;
typedef __attribute__((ext_vector_type(8))) float v8f;

#define NEG_INF (-3.402823466e38f)

// =====================================================================
// GEMM via V_WMMA_F32_16X16X4_F32 (full fp32, matches reference numerics)
// C[M x N] = A[M x K] * B (+bias).  B is [K x N] (transB=0) or [N x K] (transB=1).
// Block = 128 threads (4 wave32), tile 64(M) x 64(N), K staged in LDS by 16,
// double-buffered (regs -> LDS) with one barrier per stage.
// LDS layout: As[row][k] and Bs[n][k], both padded to stride 18 dwords so a
// (k,k+1) fragment pair is one aligned ds_load_b64 and lanes hit distinct banks.
// =====================================================================
constexpr int PK_BM = 64, PK_BN = 64, PK_BK = 16, PK_LDST = 18;

__device__ __forceinline__ void pk_load_stage(
    const float* __restrict__ A, int lda,
    const float* __restrict__ B, int ldb, int transB,
    int tile_m, int tile_n, int k0, int tid,
    float* aReg, float* bReg)
{
  #pragma unroll
  for (int j = 0; j < 8; j++) {
    int i = tid + j * 128;              // 0..1023 over 64x16 A tile
    int r = i >> 4, c = i & 15;
    aReg[j] = A[(size_t)(tile_m + r) * lda + (k0 + c)];
  }
  #pragma unroll
  for (int j = 0; j < 8; j++) {
    int i = tid + j * 128;              // 0..1023 over 16x64 B tile (k-major idx)
    int r = i >> 6, c = i & 63;         // r = k, c = n
    bReg[j] = transB ? B[(size_t)(tile_n + c) * ldb + (k0 + r)]
                     : B[(size_t)(k0 + r) * ldb + (tile_n + c)];
  }
}

__device__ __forceinline__ void pk_store_stage(
    float* __restrict__ As, float* __restrict__ Bs, int tid,
    const float* aReg, const float* bReg)
{
  #pragma unroll
  for (int j = 0; j < 8; j++) {
    int i = tid + j * 128;
    int r = i >> 4, c = i & 15;
    As[r * PK_LDST + c] = aReg[j];      // row-major, padded
  }
  #pragma unroll
  for (int j = 0; j < 8; j++) {
    int i = tid + j * 128;
    int r = i >> 6, c = i & 63;
    Bs[c * PK_LDST + r] = bReg[j];      // n-major, padded
  }
}

__global__ __launch_bounds__(128) void pk_gemm_wmma_f32(
    const float* __restrict__ A, int lda,
    const float* __restrict__ B, int ldb, int transB,
    const float* __restrict__ bias,
    float* __restrict__ C, int ldc, int K)
{
  __shared__ float As[2][PK_BM * PK_LDST];
  __shared__ float Bs[2][PK_BN * PK_LDST];

  const int tile_m = blockIdx.y * PK_BM;
  const int tile_n = blockIdx.x * PK_BN;
  const int tid  = threadIdx.x;
  const int wave = tid >> 5;
  const int lane = tid & 31;
  const int lrow  = lane & 15;          // row/col within 16
  const int khalf = (lane >> 4) * 2;    // lanes 16-31 hold K=+2,+3 (ISA A-layout)
  const int row0  = wave * 16;

  v8f acc[4] = {};
  float aReg[8], bReg[8];

  // Prologue: stage 0
  pk_load_stage(A, lda, B, ldb, transB, tile_m, tile_n, 0, tid, aReg, bReg);
  pk_store_stage(As[0], Bs[0], tid, aReg, bReg);

  int cur = 0;
  for (int k0 = 0; k0 < K; k0 += PK_BK) {
    __syncthreads();                    // buf[cur] ready; buf[cur^1] fully read
    const bool more = (k0 + PK_BK) < K;
    if (more)
      pk_load_stage(A, lda, B, ldb, transB, tile_m, tile_n, k0 + PK_BK, tid,
                    aReg, bReg);

    const float* __restrict__ as = &As[cur][0];
    const float* __restrict__ bs = &Bs[cur][0];
    #pragma unroll
    for (int kk = 0; kk < PK_BK; kk += 4) {
      v2f a = *(const v2f*)&as[(row0 + lrow) * PK_LDST + kk + khalf];
      #pragma unroll
      for (int nt = 0; nt < 4; nt++) {
        v2f b = *(const v2f*)&bs[(nt * 16 + lrow) * PK_LDST + kk + khalf];
        acc[nt] = __builtin_amdgcn_wmma_f32_16x16x4_f32(
            false, a, false, b, (short)0, acc[nt], false, false);
      }
    }

    if (more)
      pk_store_stage(As[cur ^ 1], Bs[cur ^ 1], tid, aReg, bReg);
    cur ^= 1;
  }

  // Writeback: VGPR i -> M=i (lanes 0-15) / M=8+i (lanes 16-31), N=lane%16
  const int rbase = tile_m + wave * 16 + (lane >> 4) * 8;
  #pragma unroll
  for (int nt = 0; nt < 4; nt++) {
    int col = tile_n + nt * 16 + lrow;
    float bv = bias ? bias[col] : 0.0f;
    #pragma unroll
    for (int i = 0; i < 8; i++) {
      C[(size_t)(rbase + i) * ldc + col] = acc[nt][i] + bv;
    }
  }
}

// =====================================================================
// Wave32 top-32-of-512 via iterative wave argmax (butterfly shuffles).
// Lane `sel` ends up holding selection #sel (value + global index in [0,512)).
// =====================================================================
__device__ __forceinline__ void wave_top32_of512(
    const float* __restrict__ s, int lane, float& selV, int& selI)
{
  float v[16];
  #pragma unroll
  for (int i = 0; i < 16; i++) v[i] = s[lane + 32 * i];

  unsigned taken = 0u;
  selV = NEG_INF; selI = 0;
  for (int sel = 0; sel < 32; sel++) {
    float lm = NEG_INF; int li = 0;
    #pragma unroll
    for (int i = 0; i < 16; i++) {
      bool ok = !((taken >> i) & 1u) && (v[i] > lm);
      lm = ok ? v[i] : lm;
      li = ok ? i : li;
    }
    float bm = lm; int bl = lane;
    #pragma unroll
    for (int off = 16; off > 0; off >>= 1) {
      float ov = __shfl_xor(bm, off, 32);
      int   ol = __shfl_xor(bl, off, 32);
      if (ov > bm || (ov == bm && ol < bl)) { bm = ov; bl = ol; }
    }
    if (lane == bl) taken |= (1u << li);
    int liw = __shfl(li, bl, 32);            // winner's local slot (uniform)
    if (lane == sel) { selV = bm; selI = bl + 32 * liw; }
  }
}

// One wave32 per token: top32(a), top32(b), top32 of 32x32 sums, softmax.
__global__ __launch_bounds__(256) void pk_topk_kernel(
    const float* __restrict__ scores_a,
    const float* __restrict__ scores_b,
    int*   __restrict__ out_idx,    // [T][32]
    float* __restrict__ out_w)      // [T][32]
{
  const int lane = threadIdx.x & 31;
  const int wv   = threadIdx.x >> 5;
  const int t    = blockIdx.x * 8 + wv;

  float saV, sbV; int saI, sbI;
  wave_top32_of512(scores_a + (size_t)t * 512, lane, saV, saI);
  wave_top32_of512(scores_b + (size_t)t * 512, lane, sbV, sbI);

  // Candidate grid: lane j = column j (sb_j); rows i in registers.
  float cand[32];
  #pragma unroll
  for (int i = 0; i < 32; i++)
    cand[i] = __shfl(saV, i, 32) + sbV;

  unsigned taken = 0u;
  float fsV = NEG_INF; int fsI = 0;
  for (int sel = 0; sel < 32; sel++) {
    float lm = NEG_INF; int li = 0;
    #pragma unroll
    for (int i = 0; i < 32; i++) {
      bool ok = !((taken >> i) & 1u) && (cand[i] > lm);
      lm = ok ? cand[i] : lm;
      li = ok ? i : li;
    }
    float bm = lm; int bl = lane;
    #pragma unroll
    for (int off = 16; off > 0; off >>= 1) {
      float ov = __shfl_xor(bm, off, 32);
      int   ol = __shfl_xor(bl, off, 32);
      if (ov > bm || (ov == bm && ol < bl)) { bm = ov; bl = ol; }
    }
    if (lane == bl) taken |= (1u << li);
    // winner at (row = its li, col = bl): idx = ia[li]*512 + ib[bl]
    int liw = __shfl(li, bl, 32);
    int ci  = __shfl(saI, liw, 32) * 512 + __shfl(sbI, bl, 32);
    if (lane == sel) { fsV = bm; fsI = ci; }
  }

  // Softmax over the 32 finals (one per lane)
  float mx = fsV;
  #pragma unroll
  for (int off = 16; off > 0; off >>= 1) {
    float o = __shfl_xor(mx, off, 32);
    mx = o > mx ? o : mx;
  }
  float e = __expf(fsV - mx);
  float sum = e;
  #pragma unroll
  for (int off = 16; off > 0; off >>= 1) sum += __shfl_xor(sum, off, 32);

  out_idx[(size_t)t * 32 + lane] = fsI;
  out_w  [(size_t)t * 32 + lane] = e / sum;
}

// =====================================================================
// Gather + weighted sum: out[t][:] = sum_k w_k * values[idx_k][:]
// HBM-bound phase (~512 MB). One WG (256 threads, float2 lanes) per token.
// =====================================================================
__global__ __launch_bounds__(256) void pk_gather_kernel(
    const float* __restrict__ values,
    const int*   __restrict__ idx,
    const float* __restrict__ w,
    float* __restrict__ out)
{
  const int t = blockIdx.x;
  __shared__ int   sIdx[32];
  __shared__ float sW[32];
  if (threadIdx.x < 32) {
    sIdx[threadIdx.x] = idx[(size_t)t * 32 + threadIdx.x];
    sW  [threadIdx.x] = w  [(size_t)t * 32 + threadIdx.x];
  }
  __syncthreads();

  float2 acc = make_float2(0.0f, 0.0f);
  #pragma unroll 4
  for (int k = 0; k < 32; k++) {
    const float* row = values + (size_t)sIdx[k] * 512;
    // prefetch next selected row (global_prefetch_b8 on gfx1250)
    if (k + 1 < 32)
      __builtin_prefetch(values + (size_t)sIdx[k + 1] * 512 + threadIdx.x * 2, 0, 0);
    float2 v = ((const float2*)row)[threadIdx.x];
    float  wk = sW[k];
    acc.x += wk * v.x;
    acc.y += wk * v.y;
  }
  ((float2*)(out + (size_t)t * 512))[threadIdx.x] = acc;
}

// =====================================================================
// Fused residual add + bias + LayerNorm (D=1024). One WG per token.
// =====================================================================
__device__ __forceinline__ float pk_block_sum(float v, float* lds) {
  const int lane = threadIdx.x & 31, wv = threadIdx.x >> 5;
  #pragma unroll
  for (int off = 16; off > 0; off >>= 1) v += __shfl_xor(v, off, 32);
  if (lane == 0) lds[wv] = v;
  __syncthreads();
  float r = (threadIdx.x < 8) ? lds[threadIdx.x] : 0.0f;
  if (wv == 0) {
    #pragma unroll
    for (int off = 16; off > 0; off >>= 1) r += __shfl_xor(r, off, 32);
    if (lane == 0) lds[0] = r;
  }
  __syncthreads();
  float s = lds[0];
  __syncthreads();
  return s;
}

__global__ __launch_bounds__(256) void pk_addln_kernel(
    const float* __restrict__ ymm,    // out@Wo   [T,1024]
    const float* __restrict__ x,      //          [T,1024]
    const float* __restrict__ bo,
    const float* __restrict__ gamma,
    const float* __restrict__ beta,
    float* __restrict__ out)
{
  constexpr int D = 1024;
  const int t = blockIdx.x;
  __shared__ float lds[8];

  float yv[4];
  float s = 0.0f;
  #pragma unroll
  for (int i = 0; i < 4; i++) {
    int c = threadIdx.x + i * 256;
    float v = ymm[(size_t)t * D + c] + x[(size_t)t * D + c] + bo[c];
    yv[i] = v;
    s += v;
  }
  float mu = pk_block_sum(s, lds) * (1.0f / D);

  float sq = 0.0f;
  #pragma unroll
  for (int i = 0; i < 4; i++) {
    float d = yv[i] - mu;
    sq += d * d;
  }
  float var = pk_block_sum(sq, lds) * (1.0f / D);
  float rstd = __frsqrt_rn(var + 1e-5f);

  #pragma unroll
  for (int i = 0; i < 4; i++) {
    int c = threadIdx.x + i * 256;
    out[(size_t)t * D + c] = (yv[i] - mu) * rstd * gamma[c] + beta[c];
  }
}

// =====================================================================
extern "C" void kernel_launch(void* const* d_in, const int* in_sizes, int n_in,
                              void* d_out, int out_size, void* d_ws, size_t ws_size,
                              hipStream_t stream) {
  (void)n_in; (void)out_size; (void)ws_size;
  const float* x      = (const float*)d_in[0];
  const float* Wq     = (const float*)d_in[1];
  const float* bq     = (const float*)d_in[2];
  const float* cba    = (const float*)d_in[3];
  const float* cbb    = (const float*)d_in[4];
  const float* values = (const float*)d_in[5];
  const float* Wo     = (const float*)d_in[6];
  const float* bo     = (const float*)d_in[7];
  const float* gamma  = (const float*)d_in[8];
  const float* beta   = (const float*)d_in[9];
  float* out = (float*)d_out;

  const int D = 1024;
  const int nT = in_sizes[0] / D;           // 8192 tokens

  float* ws = (float*)d_ws;
  size_t off = 0;
  float* q    = ws + off; off += (size_t)nT * 512;   // queries [T,512]
  float* sa   = ws + off; off += (size_t)nT * 512;   // scores_a
  float* sb   = ws + off; off += (size_t)nT * 512;   // scores_b
  float* ov   = ws + off; off += (size_t)nT * 512;   // weighted values [T,512]
  float* ymm  = ws + off; off += (size_t)nT * 1024;  // ov @ Wo
  int*   fidx = (int*)(ws + off); off += (size_t)nT * 32;
  float* fw   = ws + off; off += (size_t)nT * 32;

  // 1) q = x @ Wq + bq            [8192x1024]x[1024x512]
  pk_gemm_wmma_f32<<<dim3(512 / 64, nT / 64), 128, 0, stream>>>(
      x, D, Wq, 512, 0, bq, q, 512, D);

  // 2) scores_a = qa @ cba^T      [8192x256]x[256x512]
  pk_gemm_wmma_f32<<<dim3(512 / 64, nT / 64), 128, 0, stream>>>(
      q, 512, cba, 256, 1, nullptr, sa, 512, 256);

  // 3) scores_b = qb @ cbb^T
  pk_gemm_wmma_f32<<<dim3(512 / 64, nT / 64), 128, 0, stream>>>(
      q + 256, 512, cbb, 256, 1, nullptr, sb, 512, 256);

  // 4) two-level top-k + softmax (one wave32/token, 8 tokens/WG)
  pk_topk_kernel<<<nT / 8, 256, 0, stream>>>(sa, sb, fidx, fw);

  // 5) gather values + weighted sum (HBM-bound)
  pk_gather_kernel<<<nT, 256, 0, stream>>>(values, fidx, fw, ov);

  // 6) ymm = ov @ Wo              [8192x512]x[512x1024]
  pk_gemm_wmma_f32<<<dim3(1024 / 64, nT / 64), 128, 0, stream>>>(
      ov, 512, Wo, D, 0, nullptr, ymm, D, 512);

  // 7) y = LN(x + ymm + bo) * gamma + beta
  pk_addln_kernel<<<nT, 256, 0, stream>>>(ymm, x, bo, gamma, beta, out);
}